// ModulatedConv_68487548502379
// MI455X (gfx1250) — compile-verified
//
#include <hip/hip_runtime.h>

typedef __attribute__((ext_vector_type(2))) float v2f;
typedef __attribute__((ext_vector_type(8))) float v8f;

#define BSZ   16
#define HH    256
#define WW    256
#define CIN   64
#define COUT  64
#define KTOT  576   // 9 taps * 64 input channels
#define WLIN_ELEMS ((size_t)BSZ * COUT * KTOT)   // 589824 floats

// ---------------------------------------------------------------------------
// Kernel 1: per-sample modulate + demodulate the 3x3 conv weights.
// Output layout wlin[b][cout][k], k = (kh*3+kw)*64 + ci (k contiguous) so the
// main kernel's WMMA B-fragments are 8-byte contiguous LDS loads.
// Also (re)zeros the 64-float padding page used by the conv kernel.
// ---------------------------------------------------------------------------
__global__ __launch_bounds__(256) void modw_kernel(const float* __restrict__ w,
                                                   const float* __restrict__ styles,
                                                   float* __restrict__ wlin,
                                                   float* __restrict__ zpage) {
  int t = blockIdx.x * blockDim.x + threadIdx.x;
  if (t < CIN) zpage[t] = 0.0f;                 // zero padding page every call
  if (t >= BSZ * COUT) return;
  int b  = t >> 6;
  int co = t & 63;
  const float gain = 1.0f / (256.0f * 256.0f * 64.0f);  // GAIN*LR/(H*W*Cin)
  float sum = 0.0f;
  for (int k = 0; k < KTOT; ++k) {
    float s  = styles[b * CIN + (k & 63)] + 1.0f;
    float wv = w[k * COUT + co] * gain * s;   // w is [kh][kw][ci][cout]
    sum += wv * wv;
  }
  float d = rsqrtf(sum + 1e-8f);
  float* out = wlin + ((size_t)b * COUT + co) * (size_t)KTOT;
  for (int k = 0; k < KTOT; ++k) {
    float s = styles[b * CIN + (k & 63)] + 1.0f;
    out[k] = w[k * COUT + co] * gain * s * d;
  }
}

// ---------------------------------------------------------------------------
// Kernel 2: implicit-GEMM 3x3 conv using V_WMMA_F32_16X16X4_F32.
//   M = pixels, N = 64 couts (4 x 16 tiles), K = 576.
// WG = 256 threads (8 waves); each wave owns 32 pixels (two M tiles), so a WG
// covers one full 256-pixel image row of one sample. The sample's full
// demodulated weight matrix (64x576 f32 = 144 KB) lives in LDS; two WGs fit
// in the 320 KB WGP LDS for occupancy.
// SAME padding: out-of-bounds lanes redirect their A-fragment base pointer to
// a zeroed 64-float page (per-tap pointer select, unconditional inner loads).
// ---------------------------------------------------------------------------
__global__ __launch_bounds__(256) void conv_kernel(const float* __restrict__ x,
                                                   const float* __restrict__ wlin,
                                                   const float* __restrict__ zpage,
                                                   float* __restrict__ y) {
  extern __shared__ float wsm[];               // COUT*KTOT floats = 147456 B
  const int h = blockIdx.x;
  const int b = blockIdx.y;

  // Cooperative LDS stage of this sample's weights (L2-resident source).
  {
    const float4* src = (const float4*)(wlin + (size_t)b * COUT * KTOT);
    float4* dst = (float4*)wsm;
    const int n4 = COUT * KTOT / 4;            // 9216 float4s
    for (int i = threadIdx.x; i < n4; i += 256) dst[i] = src[i];
  }
  __syncthreads();

  const int lane  = threadIdx.x & 31;
  const int wid   = threadIdx.x >> 5;
  const int m     = lane & 15;                 // M index (pixel) / N index (cout)
  const int khalf = lane >> 4;                 // ISA: lanes 16-31 carry K=2,3
  const int px0   = wid * 32 + m;              // first pixel tile column
  const int px1   = px0 + 16;                  // second pixel tile column

  // B fragment base: cout row (m) in [cout][k] tile, K-half folded in.
  const float* wrow = wsm + (size_t)m * KTOT + khalf * 2;

  v8f acc[8] = {};                             // [0..3]: px0 tile, [4..7]: px1 tile

  const float* xrow = x + (size_t)b * HH * WW * CIN + khalf * 2;
  const float* zb   = zpage + khalf * 2;       // zero page, same K-half offset

#pragma unroll
  for (int dh = -1; dh <= 1; ++dh) {
    const int hy = h + dh;
    const bool vrow = (unsigned)hy < HH;
#pragma unroll
    for (int dw = -1; dw <= 1; ++dw) {
      const int wx0 = px0 + dw;
      const int wx1 = px1 + dw;
      const bool v0 = vrow && ((unsigned)wx0 < WW);
      const bool v1 = vrow && ((unsigned)wx1 < WW);
      // Per-lane pointer select: valid -> image, invalid -> zero page.
      const float* xp0 = v0 ? (xrow + ((size_t)hy * WW + wx0) * CIN) : zb;
      const float* xp1 = v1 ? (xrow + ((size_t)hy * WW + wx1) * CIN) : zb;
      const int kbase = ((dh + 1) * 3 + (dw + 1)) * 64;
#pragma unroll
      for (int c4 = 0; c4 < 16; ++c4) {        // 4 input channels per step
        v2f a0 = *(const v2f*)(xp0 + c4 * 4);  // unconditional (zero page pads)
        v2f a1 = *(const v2f*)(xp1 + c4 * 4);
        const int k = kbase + c4 * 4;
#pragma unroll
        for (int nt = 0; nt < 4; ++nt) {
          v2f bf = *(const v2f*)(wrow + (size_t)nt * 16 * KTOT + k);
          acc[nt] = __builtin_amdgcn_wmma_f32_16x16x4_f32(
              false, a0, false, bf, (short)0, acc[nt], false, false);
          acc[4 + nt] = __builtin_amdgcn_wmma_f32_16x16x4_f32(
              false, a1, false, bf, (short)0, acc[4 + nt], false, false);
        }
      }
    }
  }

  // D layout (ISA 7.12.2): VGPR v holds M=v (lanes 0-15) / M=v+8 (lanes 16-31),
  // N = lane&15. Output is NHWC [B,H,W,64].
  const int n  = lane & 15;
  const int hb = lane >> 4;
  float* ybase = y + ((size_t)b * HH + h) * (size_t)WW * COUT;
#pragma unroll
  for (int v = 0; v < 8; ++v) {
    const int p0 = wid * 32 + v + hb * 8;
    float* yp0 = ybase + (size_t)p0 * COUT + n;
    float* yp1 = yp0 + 16 * COUT;              // p1 = p0 + 16
#pragma unroll
    for (int nt = 0; nt < 4; ++nt) {
      yp0[nt * 16] = acc[nt][v];
      yp1[nt * 16] = acc[4 + nt][v];
    }
  }
}

// ---------------------------------------------------------------------------
extern "C" void kernel_launch(void* const* d_in, const int* in_sizes, int n_in,
                              void* d_out, int out_size, void* d_ws, size_t ws_size,
                              hipStream_t stream) {
  const float* x      = (const float*)d_in[0];  // [16,256,256,64]
  const float* styles = (const float*)d_in[1];  // [16,64]
  const float* w      = (const float*)d_in[2];  // [3,3,64,64]
  float* y     = (float*)d_out;                 // [16,256,256,64]
  float* wlin  = (float*)d_ws;                  // 2,359,296 B
  float* zpage = wlin + WLIN_ELEMS;             // + 256 B zero page

  (void)in_sizes; (void)n_in; (void)out_size; (void)ws_size;

  modw_kernel<<<dim3((BSZ * COUT + 255) / 256), 256, 0, stream>>>(w, styles, wlin, zpage);

  conv_kernel<<<dim3(HH, BSZ), 256,
                (size_t)(COUT * KTOT * sizeof(float)), stream>>>(x, wlin, zpage, y);
}